// CSUNReward_90434831385277
// MI455X (gfx1250) — compile-verified
//
#include <hip/hip_runtime.h>
#include <hip/hip_bf16.h>

typedef __attribute__((ext_vector_type(2))) float v2f;
typedef __attribute__((ext_vector_type(8))) float v8f;

// Coefficients from the reference (f64 hex literals, applied in f32 math by jnp)
__device__ constexpr float CE = (float)0x1.8d7d565a99f87p-1; // ~0.7764
__device__ constexpr float CA = (float)0x1.ca0aa695981e5p-3; // ~0.2237

// ---------------------------------------------------------------------------
// Kernel 1: prefix-sum of each 103-dim row, expressed as a triangular matmul
//   Out[m, d] = sum_{k<=d} In[m, k]  ==  In(16xK) @ U(KxD), U upper-tri ones
// One wave32 per 16-row block; V_WMMA_F32_16X16X4_F32 accumulation over K.
// Result is stored TRANSPOSED: outT[d * ldT + row] for coalesced reads later.
// ---------------------------------------------------------------------------
__global__ void cumsum_wmma_kernel(const float* __restrict__ in,
                                   float* __restrict__ outT, int ldT) {
  constexpr int D = 103;
  const int wave   = threadIdx.x >> 5;
  const int lane   = threadIdx.x & 31;
  const int rowBase = (blockIdx.x * 8 + wave) * 16;
  const int m      = lane & 15;          // A: M index / B: N index / D: N index
  const int khalf  = (lane >> 4) << 1;   // 0 for lanes 0-15, 2 for lanes 16-31
  const int mD     = (lane < 16) ? 0 : 8; // C/D: M offset for this lane half
  const float* rowPtr = in + (size_t)(rowBase + m) * D;

  for (int t = 0; t < 7; ++t) {          // 7 output tiles cover 112 >= 103 cols
    v8f c = {0.f, 0.f, 0.f, 0.f, 0.f, 0.f, 0.f, 0.f};
    const int kend = 4 * t + 4;          // k-steps covering kk <= 16t+15
    const int dd   = 16 * t + m;         // this lane's output column
    for (int k = 0; k < kend; ++k) {
      const int kk0 = 4 * k + khalf;
      v2f a, b;
      // A fragment (16x4 f32): lanes 0-15 hold K=0,1; lanes 16-31 hold K=2,3
      a.x = (kk0     < D) ? rowPtr[kk0]     : 0.f;
      a.y = (kk0 + 1 < D) ? rowPtr[kk0 + 1] : 0.f;
      // B fragment (4x16): upper-triangular ones, B[kk, n] = (kk <= 16t+n)
      b.x = (kk0     <= dd) ? 1.f : 0.f;
      b.y = (kk0 + 1 <= dd) ? 1.f : 0.f;
      c = __builtin_amdgcn_wmma_f32_16x16x4_f32(
          /*neg_a=*/false, a, /*neg_b=*/false, b,
          /*c_mod=*/(short)0, c, /*reuse_a=*/false, /*reuse_b=*/false);
    }
    // D tile: VGPR v holds (M = v + mD, N = m). Store transposed.
    if (dd < D) {
      float* col = outT + (size_t)dd * ldT + rowBase + mD;
#pragma unroll
      for (int v = 0; v < 8; ++v) col[v] = c[v];
    }
  }
}

// ---------------------------------------------------------------------------
// Kernel 2: simple transpose (rows x cols) -> (cols x rows), for amd embs
// ---------------------------------------------------------------------------
__global__ void transpose_kernel(const float* __restrict__ in,
                                 float* __restrict__ outT, int rows, int cols) {
  const int idx = blockIdx.x * blockDim.x + threadIdx.x;
  if (idx < rows * cols) {
    const int r = idx / cols;
    const int c = idx - r * cols;
    outT[(size_t)c * rows + r] = in[idx];
  }
}

// ---------------------------------------------------------------------------
// Kernel 3: init novelty accumulators to +inf (as ordered uint bits)
// ---------------------------------------------------------------------------
__global__ void init_nov_kernel(unsigned* __restrict__ nov) {
  nov[threadIdx.x] = 0x7F800000u; // +inf
}

// ---------------------------------------------------------------------------
// Kernel 4: uniqueness — one block per gen row i; 256 threads over j
// ---------------------------------------------------------------------------
__global__ void uniqueness_kernel(const float* __restrict__ GcT,
                                  const float* __restrict__ GaT,
                                  float* __restrict__ uni) {
  constexpr int N = 256, DE = 103, DA = 100;
  __shared__ float sE[DE];
  __shared__ float sA[DA];
  __shared__ float red[N];
  const int i = blockIdx.x;
  const int tid = threadIdx.x;
  if (tid < DE) sE[tid] = GcT[tid * N + i];
  if (tid < DA) sA[tid] = GaT[tid * N + i];
  __syncthreads();
  float d1 = 0.f, d2 = 0.f;
  for (int d = 0; d < DE; ++d) d1 += fabsf(GcT[d * N + tid] - sE[d]); // L1
  for (int d = 0; d < DA; ++d) d2 = fmaxf(d2, fabsf(GaT[d * N + tid] - sA[d])); // Cheb
  red[tid] = (CE * d1) / (1.f + d1) + (CA * d2) / (1.f + d2);
  __syncthreads();
  for (int s = N / 2; s > 0; s >>= 1) {
    if (tid < s) red[tid] += red[tid + s];
    __syncthreads();
  }
  if (tid == 0) uni[i] = red[0] / (float)(N - 1);
}

// ---------------------------------------------------------------------------
// Kernel 5: novelty — 8 gen rows per block (LDS-staged, 8x ref reuse),
// 256 threads over a 256-wide j chunk; tree-min + uint atomicMin.
// ---------------------------------------------------------------------------
__global__ void novelty_kernel(const float* __restrict__ GcT,
                               const float* __restrict__ GaT,
                               const float* __restrict__ RcT,
                               const float* __restrict__ RaT,
                               unsigned* __restrict__ nov) {
  constexpr int N = 256, M = 4096, DE = 103, DA = 100, IT = 8;
  __shared__ float sE[IT][DE];
  __shared__ float sA[IT][DA];
  __shared__ float red[256];
  const int tid = threadIdx.x;
  const int ibase = blockIdx.x * IT;
  const int j = blockIdx.y * blockDim.x + tid;

  for (int idx = tid; idx < IT * DE; idx += 256) {
    const int ii = idx / DE, d = idx - ii * DE;
    sE[ii][d] = GcT[(size_t)d * N + ibase + ii];
  }
  for (int idx = tid; idx < IT * DA; idx += 256) {
    const int ii = idx / DA, d = idx - ii * DA;
    sA[ii][d] = GaT[(size_t)d * N + ibase + ii];
  }
  __syncthreads();

  float d1[IT], d2[IT];
#pragma unroll
  for (int ii = 0; ii < IT; ++ii) { d1[ii] = 0.f; d2[ii] = 0.f; }

  for (int d = 0; d < DE; ++d) {
    const float r = RcT[(size_t)d * M + j]; // coalesced across lanes
#pragma unroll
    for (int ii = 0; ii < IT; ++ii) d1[ii] += fabsf(r - sE[ii][d]);
  }
  for (int d = 0; d < DA; ++d) {
    const float r = RaT[(size_t)d * M + j];
#pragma unroll
    for (int ii = 0; ii < IT; ++ii) d2[ii] = fmaxf(d2[ii], fabsf(r - sA[ii][d]));
  }

  for (int ii = 0; ii < IT; ++ii) {
    red[tid] = (CE * d1[ii]) / (1.f + d1[ii]) + (CA * d2[ii]) / (1.f + d2[ii]);
    __syncthreads();
    for (int s = 128; s > 0; s >>= 1) {
      if (tid < s) red[tid] = fminf(red[tid], red[tid + s]);
      __syncthreads();
    }
    // distances are >= 0, so float ordering == uint ordering of the bits
    if (tid == 0) atomicMin(nov + ibase + ii, __float_as_uint(red[0]));
    __syncthreads();
  }
}

// ---------------------------------------------------------------------------
// Kernel 6: finalize — stability clip (NaN->0) * uni * nov, NaN->0
// ---------------------------------------------------------------------------
__global__ void finalize_kernel(const float* __restrict__ uni,
                                const unsigned* __restrict__ nov,
                                const float* __restrict__ eah,
                                float* __restrict__ out) {
  const int i = threadIdx.x;
  const float e = eah[i];
  float stab = (e != e) ? 0.f
                        : fminf(fmaxf(1.f - e / 0.4289f, 0.f), 1.f);
  float s = stab * uni[i] * __uint_as_float(nov[i]);
  if (s != s) s = 0.f;
  out[i] = s;
}

// ---------------------------------------------------------------------------
extern "C" void kernel_launch(void* const* d_in, const int* in_sizes, int n_in,
                              void* d_out, int out_size, void* d_ws, size_t ws_size,
                              hipStream_t stream) {
  (void)in_sizes; (void)n_in; (void)out_size; (void)ws_size;
  const float* genE = (const float*)d_in[0]; // 256 x 103
  const float* genA = (const float*)d_in[1]; // 256 x 100
  const float* refE = (const float*)d_in[2]; // 4096 x 103
  const float* refA = (const float*)d_in[3]; // 4096 x 100
  const float* eah  = (const float*)d_in[4]; // 256
  float* out = (float*)d_out;                // 256

  // Workspace layout (floats): ~3.4 MB total
  float* ws  = (float*)d_ws;
  float* GcT = ws;                    // 103 x 256  (cumsum'd gen elmd, T)
  float* RcT = GcT + 103 * 256;       // 103 x 4096 (cumsum'd ref elmd, T)
  float* GaT = RcT + 103 * 4096;      // 100 x 256  (gen amd, T)
  float* RaT = GaT + 100 * 256;       // 100 x 4096 (ref amd, T)
  float* uni = RaT + 100 * 4096;      // 256
  unsigned* nov = (unsigned*)(uni + 256); // 256

  // 1) prefix-sum via WMMA triangular matmul (stores transposed)
  cumsum_wmma_kernel<<<256 / 128, 256, 0, stream>>>(genE, GcT, 256);
  cumsum_wmma_kernel<<<4096 / 128, 256, 0, stream>>>(refE, RcT, 4096);
  // 2) transpose amd embeddings for coalesced pairwise reads
  transpose_kernel<<<(256 * 100 + 255) / 256, 256, 0, stream>>>(genA, GaT, 256, 100);
  transpose_kernel<<<(4096 * 100 + 255) / 256, 256, 0, stream>>>(refA, RaT, 4096, 100);
  // 3) novelty accumulators to +inf
  init_nov_kernel<<<1, 256, 0, stream>>>(nov);
  // 4) uniqueness (256 x 256 pairwise, row-mean)
  uniqueness_kernel<<<256, 256, 0, stream>>>(GcT, GaT, uni);
  // 5) novelty (256 x 4096 pairwise, row-min)
  novelty_kernel<<<dim3(32, 16), 256, 0, stream>>>(GcT, GaT, RcT, RaT, nov);
  // 6) combine with stability
  finalize_kernel<<<1, 256, 0, stream>>>(uni, nov, eah, out);
}